// DynamicGraphConstruction_86088324481261
// MI455X (gfx1250) — compile-verified
//
#include <hip/hip_runtime.h>
#include <hip/hip_bf16.h>

typedef __attribute__((ext_vector_type(2))) float v2f;
typedef __attribute__((ext_vector_type(8))) float v8f;

#define D_DIM 128          // embedding dim, fixed by reference
#define BN_EPS 1e-5

// ---------------------------------------------------------------------------
// Kernel 1: per-wave tile of 16 edges. Compute diff tile A (16 x 4 per step),
// accumulate Gram = A * A^T with v_wmma_f32_16x16x4_f32 (A and A^T share the
// same VGPR layout for this shape, so SRC0 == SRC1). Diagonal of the Gram
// matrix after 32 K-steps = squared distances. Then likelihood = -log(d2),
// wave+block reduce (sum, sum^2) -> f64 atomics.
// ---------------------------------------------------------------------------
__global__ void __launch_bounds__(256)
k_dist_wmma(const float* __restrict__ src,
            const float* __restrict__ dst,
            const long long* __restrict__ graph,
            float* __restrict__ lik_out,      // logits region used as temp
            double* __restrict__ acc,         // [0]=sum, [1]=sumsq
            long long E)
{
    const int lane = threadIdx.x & 31;
    const int wave = threadIdx.x >> 5;
    const long long base = ((long long)blockIdx.x * 8 + wave) * 16;

    const int m = lane & 15;          // edge-in-tile this lane feeds
    const int h = lane >> 4;          // 0 -> local K 0,1 ; 1 -> local K 2,3

    long long e = base + m;
    if (e >= E) e = E - 1;            // clamp (selects, no EXEC divergence)
    const long long si = graph[e];
    const long long di = graph[E + e];
    const float2* __restrict__ sp = (const float2*)(src + si * D_DIM);
    const float2* __restrict__ dp = (const float2*)(dst + di * D_DIM);

    v8f c = {};
#pragma unroll 8
    for (int kb = 0; kb < D_DIM / 4; ++kb) {
        const int idx = kb * 2 + h;   // float2 index: global k = 4*kb + 2*h
        float2 s2 = sp[idx];
        float2 t2 = dp[idx];
        v2f a;
        a.x = s2.x - t2.x;
        a.y = s2.y - t2.y;
        // D = A * A^T + C  -> Gram matrix of the diff tile
        c = __builtin_amdgcn_wmma_f32_16x16x4_f32(false, a, false, a,
                                                  (short)0, c, false, false);
    }

    // Diagonal extraction (C layout: v[j]: lanes0-15 -> M=j, N=lane;
    // lanes16-31 -> M=j+8, N=lane-16):
    //   diag(0..7)  lives in lane L (L<8)      at c[L]
    //   diag(8..15) lives in lane L (L>=24)    at c[L-24]
    const bool active = (lane < 8) || (lane >= 24);
    int sel = ((lane < 8) ? lane : (lane - 24)) & 7;
    float d2 = c[0];
    d2 = (sel == 1) ? c[1] : d2;
    d2 = (sel == 2) ? c[2] : d2;
    d2 = (sel == 3) ? c[3] : d2;
    d2 = (sel == 4) ? c[4] : d2;
    d2 = (sel == 5) ? c[5] : d2;
    d2 = (sel == 6) ? c[6] : d2;
    d2 = (sel == 7) ? c[7] : d2;

    const long long myE = base + ((lane < 8) ? lane : (lane - 16));
    float lik = 0.f, lik2 = 0.f;
    if (active && myE < E) {
        float dd = fmaxf(d2, 1e-12f);
        float l  = -__logf(dd);
        lik_out[myE] = l;
        lik  = l;
        lik2 = l * l;
    }

    // wave32 reduction
#pragma unroll
    for (int off = 16; off > 0; off >>= 1) {
        lik  += __shfl_down(lik,  off, 32);
        lik2 += __shfl_down(lik2, off, 32);
    }
    __shared__ float sh1[8], sh2[8];
    if (lane == 0) { sh1[wave] = lik; sh2[wave] = lik2; }
    __syncthreads();
    if (threadIdx.x == 0) {
        float t1 = 0.f, t2 = 0.f;
#pragma unroll
        for (int i = 0; i < 8; ++i) { t1 += sh1[i]; t2 += sh2[i]; }
        __hip_atomic_fetch_add(&acc[0], (double)t1, __ATOMIC_RELAXED, __HIP_MEMORY_SCOPE_AGENT);
        __hip_atomic_fetch_add(&acc[1], (double)t2, __ATOMIC_RELAXED, __HIP_MEMORY_SCOPE_AGENT);
    }
}

// ---------------------------------------------------------------------------
// Kernel 2: batch-norm -> logits (in place over lik), sigmoid -> out_w,
// copy graph to output, reduce sum(sigmoid) -> acc[2].
// gmode: 0 = skip graph copy, 1 = cast to f32, 2 = raw int64.
// ---------------------------------------------------------------------------
__global__ void __launch_bounds__(256)
k_bn_sigmoid(const long long* __restrict__ graph,
             const float* __restrict__ bn_w,
             const float* __restrict__ bn_b,
             float* __restrict__ out_gf,
             long long* __restrict__ out_gi,
             int gmode,
             float* __restrict__ out_w,
             float* __restrict__ logits,     // in: likelihood, out: logits
             double* __restrict__ acc,
             long long E)
{
    const long long i = (long long)blockIdx.x * blockDim.x + threadIdx.x;

    const double mu  = acc[0] / (double)E;
    double var = acc[1] / (double)E - mu * mu;
    if (var < 0.0) var = 0.0;
    const float inv_std = (float)(1.0 / sqrt(var + BN_EPS));
    const float w = bn_w[0], b = bn_b[0];
    const float muf = (float)mu;

    float sig = 0.f;
    if (i < E) {
        const long long g0 = graph[i];
        const long long g1 = graph[E + i];
        if (gmode == 1)      { out_gf[i] = (float)g0; out_gf[E + i] = (float)g1; }
        else if (gmode == 2) { out_gi[i] = g0;        out_gi[E + i] = g1; }
        const float l = logits[i];
        const float z = w * (l - muf) * inv_std + b;
        logits[i] = z;
        sig = 1.f / (1.f + __expf(-z));
        out_w[i] = sig;
    }

#pragma unroll
    for (int off = 16; off > 0; off >>= 1)
        sig += __shfl_down(sig, off, 32);
    __shared__ float sh[8];
    const int lane = threadIdx.x & 31;
    const int wave = threadIdx.x >> 5;
    if (lane == 0) sh[wave] = sig;
    __syncthreads();
    if (threadIdx.x == 0) {
        float t = 0.f;
#pragma unroll
        for (int k = 0; k < 8; ++k) t += sh[k];
        __hip_atomic_fetch_add(&acc[2], (double)t, __ATOMIC_RELAXED, __HIP_MEMORY_SCOPE_AGENT);
    }
}

// ---------------------------------------------------------------------------
// Kernel 3: edge_weights /= mean(edge_weights)
// ---------------------------------------------------------------------------
__global__ void __launch_bounds__(256)
k_normalize(float* __restrict__ out_w, const double* __restrict__ acc, long long E)
{
    const long long i = (long long)blockIdx.x * blockDim.x + threadIdx.x;
    if (i < E) {
        const float s = (float)((double)E / acc[2]);
        out_w[i] *= s;
    }
}

extern "C" void kernel_launch(void* const* d_in, const int* in_sizes, int n_in,
                              void* d_out, int out_size, void* d_ws, size_t ws_size,
                              hipStream_t stream) {
    const float*     src   = (const float*)d_in[0];
    const float*     dst   = (const float*)d_in[1];
    const long long* graph = (const long long*)d_in[2];
    const float*     bn_w  = (const float*)d_in[3];
    const float*     bn_b  = (const float*)d_in[4];

    const long long E = (long long)in_sizes[2] / 2;
    double* acc = (double*)d_ws;
    hipMemsetAsync(d_ws, 0, 3 * sizeof(double), stream);

    // Output layout: [graph passthrough][edge_weights (E)][logits (E)].
    // Infer graph region encoding from out_size.
    long long graph_slots = (long long)out_size - 2 * E;   // f32-unit slots
    int gmode;
    if (graph_slots == 4 * E)      gmode = 2;   // raw int64 (2 slots each)
    else if (graph_slots == 2 * E) gmode = 1;   // cast to f32
    else { gmode = 0; if (graph_slots < 0) graph_slots = 0; }

    float*     out_base = (float*)d_out;
    float*     out_gf   = (gmode == 1) ? out_base : nullptr;
    long long* out_gi   = (gmode == 2) ? (long long*)d_out : nullptr;
    float*     out_w    = out_base + graph_slots;
    float*     out_lg   = out_w + E;

    const dim3 blk(256);
    const unsigned nb_tile = (unsigned)((E + 127) / 128);   // 8 waves x 16 edges
    const unsigned nb_elt  = (unsigned)((E + 255) / 256);

    k_dist_wmma<<<nb_tile, blk, 0, stream>>>(src, dst, graph, out_lg, acc, E);
    k_bn_sigmoid<<<nb_elt, blk, 0, stream>>>(graph, bn_w, bn_b,
                                             out_gf, out_gi, gmode,
                                             out_w, out_lg, acc, E);
    k_normalize<<<nb_elt, blk, 0, stream>>>(out_w, acc, E);
}